// DGSRLayers_3839700762811
// MI455X (gfx1250) — compile-verified
//
#include <hip/hip_runtime.h>
#include <math.h>

// ---------------------------------------------------------------------------
// DGSR layer for MI455X (gfx1250, wave32).
//   Stage 1: uh = user_feat @ W_u ; ih = item_feat @ W_i      (WMMA f32 GEMM)
//   Stage 2: per-node time-rank attention reduce -> cat [N,128]
//            (async-to-LDS mailbox gather + LDS compute)
//   Stage 3: out = elu(cat @ Wg + feat)                        (WMMA f32 GEMM)
// ---------------------------------------------------------------------------

typedef __attribute__((ext_vector_type(2))) float v2f;
typedef __attribute__((ext_vector_type(8))) float v8f;

#define DIM 64
#define MBL 50   // mailbox length (LU == LI == 50)

// ---------------------------------------------------------------------------
// WMMA fp32 GEMM: C[M,64] = A[M,K] @ B[K,64], optional fused residual + ELU.
// Block = 256 threads = 8 waves; each wave computes a 16x64 output slab with
// four 16x16 accumulators of V_WMMA_F32_16X16X4_F32 (exact fp32 matrix math).
//
// B is staged in LDS PRE-SWIZZLED for the WMMA B-fragment layout:
//   Bs[(k>>2)*256 + n*4 + (k&3)] = B[k][n]
// so a lane's fragment pair (B[k2][n], B[k2+1][n]) is one contiguous,
// 8B-aligned ds_load_b64, and lanes 0-15 / 16-31 hit disjoint banks
// (words 4n vs 4n+2) -> conflict-free, no repacking movs.
// ---------------------------------------------------------------------------
__global__ __launch_bounds__(256) void dgsr_wmma_gemm_f32(
    const float* __restrict__ A,    // [M,K] row-major
    const float* __restrict__ B,    // [K,64] row-major
    const float* __restrict__ Res,  // [M,64] residual or nullptr
    float* __restrict__ Cout,       // [M,64]
    int M, int K, int fuse_elu)
{
    extern __shared__ float Bs[];          // K*64 floats, swizzled
    const int tid  = threadIdx.x;
    const int wave = tid >> 5;             // 0..7
    const int lane = tid & 31;
    const int ln   = lane & 15;            // N index within tile / M row lane
    const int kh   = lane >> 4;            // half-wave K selector

    // cooperative swizzled stage of B into LDS
    for (int i = tid; i < K * 64; i += 256) {
        const int r = i >> 6, c = i & 63;  // B[r][c]
        Bs[(r >> 2) * 256 + c * 4 + (r & 3)] = B[i];
    }
    __syncthreads();

    const int row0 = blockIdx.x * 128 + wave * 16;
    int arow = row0 + ln;
    if (arow >= M) arow = M - 1;           // clamp for safe loads; store guarded
    const float* Arow = A + (size_t)arow * K;

    v8f c0 = {}, c1 = {}, c2 = {}, c3 = {};
    for (int kb = 0; kb < K; kb += 4) {
        const int k2 = kb + 2 * kh;        // lanes 0-15: K={kb,kb+1}; 16-31: {kb+2,kb+3}
        v2f a;
        a.x = Arow[k2];
        a.y = Arow[k2 + 1];
        const float* bchunk = Bs + (kb >> 2) * 256 + kh * 2;
        v2f b;
        b = *(const v2f*)(bchunk + (ln)      * 4);
        c0 = __builtin_amdgcn_wmma_f32_16x16x4_f32(false, a, false, b, (short)0, c0, false, false);
        b = *(const v2f*)(bchunk + (16 + ln) * 4);
        c1 = __builtin_amdgcn_wmma_f32_16x16x4_f32(false, a, false, b, (short)0, c1, false, false);
        b = *(const v2f*)(bchunk + (32 + ln) * 4);
        c2 = __builtin_amdgcn_wmma_f32_16x16x4_f32(false, a, false, b, (short)0, c2, false, false);
        b = *(const v2f*)(bchunk + (48 + ln) * 4);
        c3 = __builtin_amdgcn_wmma_f32_16x16x4_f32(false, a, false, b, (short)0, c3, false, false);
    }

    // C layout: VGPR r -> M = row0 + r + 8*kh ; N = nt*16 + ln
    v8f acc[4] = {c0, c1, c2, c3};
    for (int nt = 0; nt < 4; ++nt) {
        for (int r = 0; r < 8; ++r) {
            const int row = row0 + kh * 8 + r;
            if (row < M) {
                const int col = nt * 16 + ln;
                float v = acc[nt][r];
                if (Res) v += Res[(size_t)row * 64 + col];
                if (fuse_elu) v = (v > 0.f) ? v : expm1f(v);
                Cout[(size_t)row * 64 + col] = v;
            }
        }
    }
}

// ---------------------------------------------------------------------------
// Per-node DGSR attention reduce. One 64-thread block (2 waves) per node.
// Mailbox (50x64 fp32) staged in LDS with row stride 65 (conflict-free for
// both the row-walk dot phase and the column-walk accumulate phase).
// Gather uses CDNA5 GLOBAL_LOAD_ASYNC_TO_LDS_B32 (one 256B row per issue,
// tracked by ASYNCcnt; no VGPR round-trip).
// Emits cat = [h_long | h_short] into [N,128].
// ---------------------------------------------------------------------------
__global__ __launch_bounds__(64) void dgsr_attn(
    const float* __restrict__ src_h,  // [Nsrc,64] gathered-from table
    const float* __restrict__ dst_h,  // [N,64]
    const int*   __restrict__ nbr,    // [N,50]
    const int*   __restrict__ tim,    // [N,50]
    const float* __restrict__ te,     // [50,64] time encoding (query side)
    const float* __restrict__ te_k,   // [50,64] time encoding (value side)
    float* __restrict__ cat_out,      // [N,128]
    int N)
{
    __shared__ float mb[MBL][DIM + 1];
    __shared__ float h[DIM];
    __shared__ float e[MBL], alpha[MBL], e1[MBL], a1[MBL];
    __shared__ int   ts[MBL], reo[MBL];
    __shared__ int   last_s;

    const int node = blockIdx.x;
    const int t = threadIdx.x;              // 0..63
    if (node >= N) return;

    h[t] = dst_h[(size_t)node * DIM + t];
    if (t < MBL) ts[t] = tim[(size_t)node * MBL + t];

    // gather mailbox rows (index identical across the block -> coalesced 256B)
#if __has_builtin(__builtin_amdgcn_global_load_async_to_lds_b32)
    {
        typedef __attribute__((address_space(1))) int gint;
        typedef __attribute__((address_space(3))) int lint;
        for (int l = 0; l < MBL; ++l) {
            const int src = nbr[(size_t)node * MBL + l];
            __builtin_amdgcn_global_load_async_to_lds_b32(
                (gint*)(src_h + (size_t)src * DIM + t),
                (lint*)&mb[l][t], 0, 0);
        }
    }
#if __has_builtin(__builtin_amdgcn_s_wait_asynccnt)
    __builtin_amdgcn_s_wait_asynccnt(0);
#else
    asm volatile("s_wait_asynccnt 0x0" ::: "memory");
#endif
#else
    for (int l = 0; l < MBL; ++l) {
        const int src = nbr[(size_t)node * MBL + l];
        mb[l][t] = src_h[(size_t)src * DIM + t];
    }
#endif
    __syncthreads();

    if (t == 0) {                           // argmax (first occurrence)
        int best = ts[0], bi = 0;
        for (int l = 1; l < MBL; ++l)
            if (ts[l] > best) { best = ts[l]; bi = l; }
        last_s = bi;
    }
    __syncthreads();

    if (t < MBL) {
        // stable ascending rank == argsort(argsort(time)); re_o = L-1-rank
        const int tl = ts[t];
        int rank = 0;
        for (int m = 0; m < MBL; ++m) {
            const int tm = ts[m];
            rank += (tm < tl) || ((tm == tl) && (m < t));
        }
        const int ro = MBL - 1 - rank;
        reo[t] = ro;
        const float* terow = te + ro * DIM;
        float acc = 0.f;
        for (int d = 0; d < DIM; ++d)
            acc += (terow[d] + mb[t][d]) * h[d];
        e[t] = acc * 0.125f;                // 1/sqrt(64)

        const int last = last_s;
        float acc1 = 0.f;
        for (int d = 0; d < DIM; ++d)
            acc1 += mb[last][d] * mb[t][d]; // mb[last][d] broadcasts
        e1[t] = acc1 * 0.125f;
    }
    __syncthreads();

    // two serial softmaxes over L=50, one per wave (run concurrently)
    if (t == 0) {
        float m = e[0];
        for (int l = 1; l < MBL; ++l) m = fmaxf(m, e[l]);
        float s = 0.f;
        for (int l = 0; l < MBL; ++l) { float x = __expf(e[l] - m); alpha[l] = x; s += x; }
        const float inv = 1.f / s;
        for (int l = 0; l < MBL; ++l) alpha[l] *= inv;
    } else if (t == 32) {
        float m = e1[0];
        for (int l = 1; l < MBL; ++l) m = fmaxf(m, e1[l]);
        float s = 0.f;
        for (int l = 0; l < MBL; ++l) { float x = __expf(e1[l] - m); a1[l] = x; s += x; }
        const float inv = 1.f / s;
        for (int l = 0; l < MBL; ++l) a1[l] *= inv;
    }
    __syncthreads();

    // column-walk accumulate: thread t owns feature dim t
    float hl = 0.f, hs = 0.f;
    for (int l = 0; l < MBL; ++l) {
        const float a = alpha[l];
        const float b = a1[l];
        const float v = mb[l][t];
        hl += a * (v + te_k[reo[l] * DIM + t]);   // te_k row hot in L2
        hs += b * v;
    }
    cat_out[(size_t)node * 128 + t]      = hl;
    cat_out[(size_t)node * 128 + 64 + t] = hs;
}

// ---------------------------------------------------------------------------
extern "C" void kernel_launch(void* const* d_in, const int* in_sizes, int n_in,
                              void* d_out, int out_size, void* d_ws, size_t ws_size,
                              hipStream_t stream) {
    const float* user_feat = (const float*)d_in[0];
    const float* item_feat = (const float*)d_in[1];
    const float* W_u       = (const float*)d_in[2];
    const float* W_i       = (const float*)d_in[3];
    const float* Wg_u      = (const float*)d_in[4];   // [128,64]
    const float* Wg_i      = (const float*)d_in[5];   // [128,64]
    const float* i_te      = (const float*)d_in[6];
    const float* i_te_k    = (const float*)d_in[7];
    const float* u_te      = (const float*)d_in[8];
    const float* u_te_k    = (const float*)d_in[9];
    const int*   item_nbr  = (const int*)d_in[10];
    const int*   item_time = (const int*)d_in[11];
    const int*   user_nbr  = (const int*)d_in[12];
    const int*   user_time = (const int*)d_in[13];

    const int NU = in_sizes[0] / DIM;
    const int NI = in_sizes[1] / DIM;

    float* ws    = (float*)d_ws;
    float* uh    = ws;                         // [NU,64]
    float* ih    = uh + (size_t)NU * DIM;      // [NI,64]
    float* cat_u = ih + (size_t)NI * DIM;      // [NU,128]
    float* cat_i = cat_u + (size_t)NU * 128;   // [NI,128]

    float* user_out = (float*)d_out;                    // [NU,64]
    float* item_out = user_out + (size_t)NU * DIM;      // [NI,64]

    const dim3 blk(256);
    const size_t shm64  = (size_t)64  * 64 * sizeof(float);
    const size_t shm128 = (size_t)128 * 64 * sizeof(float);

    // Stage 1: node-feature GEMMs
    dgsr_wmma_gemm_f32<<<dim3((NU + 127) / 128), blk, shm64, stream>>>(
        user_feat, W_u, nullptr, uh, NU, 64, 0);
    dgsr_wmma_gemm_f32<<<dim3((NI + 127) / 128), blk, shm64, stream>>>(
        item_feat, W_i, nullptr, ih, NI, 64, 0);

    // Stage 2: attention reduces (item side gathers user mailboxes; user side
    // gathers item mailboxes). Gather tables are L2-resident (<13 MB).
    dgsr_attn<<<dim3(NI), dim3(64), 0, stream>>>(
        uh, ih, item_nbr, item_time, i_te, i_te_k, cat_i, NI);
    dgsr_attn<<<dim3(NU), dim3(64), 0, stream>>>(
        ih, uh, user_nbr, user_time, u_te, u_te_k, cat_u, NU);

    // Stage 3: output projections with fused residual + ELU
    dgsr_wmma_gemm_f32<<<dim3((NU + 127) / 128), blk, shm128, stream>>>(
        cat_u, Wg_u, user_feat, user_out, NU, 128, 1);
    dgsr_wmma_gemm_f32<<<dim3((NI + 127) / 128), blk, shm128, stream>>>(
        cat_i, Wg_i, item_feat, item_out, NI, 128, 1);
}